// SWINClassifier_74199855005910
// MI455X (gfx1250) — compile-verified
//
#include <hip/hip_runtime.h>
#include <cmath>

typedef float v2f __attribute__((ext_vector_type(2)));
typedef float v8f __attribute__((ext_vector_type(8)));

#define LN_EPS 1e-5f

// ---------------------------------------------------------------------------
// Generic batched/strided WMMA GEMM:  C = alpha * A @ B(^T) + bias  (opt +=)
// One wave (32 threads) computes a 16x16 tile of C using v_wmma_f32_16x16x4_f32.
// Branchless inner loop: out-of-range lanes read from a zeroed pad buffer
// (pointer select hoisted out of the loop), so all loads are unconditional.
// zpad must hold >= 4096 zero floats (max K here is 3072).
// ---------------------------------------------------------------------------
__global__ void __launch_bounds__(32)
k_gemm_wmma(const float* __restrict__ A, const float* __restrict__ B,
            float* __restrict__ C, const float* __restrict__ bias,
            const float* __restrict__ zpad,
            int M, int N, int K,
            long lda, long ldb, long ldc,
            long sA1, long sA2, long sB1, long sB2, long sC1, long sC2,
            int batch2, float alpha, int transB, int accum)
{
    int bz = blockIdx.z;
    int b1 = bz / batch2, b2 = bz % batch2;
    const float* Ab = A + (long)b1 * sA1 + (long)b2 * sA2;
    const float* Bb = B + (long)b1 * sB1 + (long)b2 * sB2;
    float*       Cb = C + (long)b1 * sC1 + (long)b2 * sC2;

    int lane = threadIdx.x;
    int half = lane >> 4;      // 0: lanes 0-15, 1: lanes 16-31
    int lid  = lane & 15;
    int i0 = blockIdx.y << 4;
    int j0 = blockIdx.x << 4;

    int mA = i0 + lid;  bool mok = (mA < M);
    int nB = j0 + lid;  bool nok = (nB < N);

    // branchless operand base pointers (invalid lanes -> zero pad)
    const float* aptr = mok ? (Ab + (long)mA * lda) : zpad;
    const float* bptr;
    long bstr;
    if (transB) { bptr = nok ? (Bb + (long)nB * ldb) : zpad; bstr = 1; }
    else        { bptr = nok ? (Bb + nB)             : zpad; bstr = nok ? ldb : 0; }

    v8f acc = {0.f,0.f,0.f,0.f,0.f,0.f,0.f,0.f};
    int K4 = K & ~3;
    int kh = half << 1;        // this lane's K offset within the 4-step

    for (int k0 = 0; k0 < K4; k0 += 4) {
        int ka = k0 + kh;
        v2f a, b;
        a.x = aptr[ka];
        a.y = aptr[ka + 1];
        b.x = bptr[(long)ka * bstr];
        b.y = bptr[(long)(ka + 1) * bstr];
        acc = __builtin_amdgcn_wmma_f32_16x16x4_f32(false, a, false, b,
                                                    (short)0, acc, false, false);
    }
    if (K4 < K) {              // uniform tail (K % 4 != 0), clamped + selected
        int ka  = K4 + kh;
        int k0i = (ka     < K) ? ka     : 0;
        int k1i = (ka + 1 < K) ? ka + 1 : 0;
        float a0 = aptr[k0i], a1 = aptr[k1i];
        float b0 = bptr[(long)k0i * bstr], b1v = bptr[(long)k1i * bstr];
        v2f a, b;
        a.x = (ka     < K) ? a0  : 0.f;
        a.y = (ka + 1 < K) ? a1  : 0.f;
        b.x = (ka     < K) ? b0  : 0.f;
        b.y = (ka + 1 < K) ? b1v : 0.f;
        acc = __builtin_amdgcn_wmma_f32_16x16x4_f32(false, a, false, b,
                                                    (short)0, acc, false, false);
    }

    const float* bp = (bias != nullptr && nok) ? (bias + nB) : zpad;
    float bv = *bp;

    bool full = (i0 + 16 <= M) && (j0 + 16 <= N);
    int mbase = i0 + (half << 3);
    if (full) {
        if (accum) {
#pragma unroll
            for (int v = 0; v < 8; ++v)
                Cb[(long)(mbase + v) * ldc + nB] += alpha * acc[v] + bv;
        } else {
#pragma unroll
            for (int v = 0; v < 8; ++v)
                Cb[(long)(mbase + v) * ldc + nB] = alpha * acc[v] + bv;
        }
    } else {
#pragma unroll
        for (int v = 0; v < 8; ++v) {
            int m = mbase + v;
            if (m < M && nok) {
                float r = alpha * acc[v] + bv;
                long o = (long)m * ldc + nB;
                if (accum) Cb[o] += r; else Cb[o] = r;
            }
        }
    }
}

// ---------------------------------------------------------------------------
// LayerNorm over last dim C; one wave per row. g/b may be null (no affine).
// ---------------------------------------------------------------------------
__global__ void __launch_bounds__(32)
k_layernorm(const float* __restrict__ x, float* __restrict__ y,
            const float* __restrict__ g, const float* __restrict__ b,
            int T, int C)
{
    int row = blockIdx.x;
    if (row >= T) return;
    const float* xr = x + (long)row * C;
    float s = 0.f, ss = 0.f;
    for (int c = threadIdx.x; c < C; c += 32) { float v = xr[c]; s += v; ss += v * v; }
    for (int o = 16; o; o >>= 1) { s += __shfl_down(s, o); ss += __shfl_down(ss, o); }
    s = __shfl(s, 0); ss = __shfl(ss, 0);
    float mean = s / C;
    float var  = ss / C - mean * mean;
    float inv  = rsqrtf(var + LN_EPS);
    float* yr = y + (long)row * C;
    for (int c = threadIdx.x; c < C; c += 32) {
        float v = (xr[c] - mean) * inv;
        if (g) v = v * g[c] + b[c];
        yr[c] = v;
    }
}

// ---------------------------------------------------------------------------
// Window partition with pad + cyclic shift folded into the gather.
// shifted[z] = padded[(z+s) mod Dp]; pad region -> 0.
// ---------------------------------------------------------------------------
__global__ void k_win_gather(const float* __restrict__ y, float* __restrict__ xw,
                             int D, int H, int W, int C,
                             int wd, int wh, int ww,
                             int gd, int gh, int gw,
                             int sd, int sh, int sw,
                             int Dp, int Hp, int Wp)
{
    long n = (long)wd * wh * ww;
    long total = (long)gd * gh * gw * n * C;
    long idx = blockIdx.x * (long)blockDim.x + threadIdx.x;
    if (idx >= total) return;
    int c = (int)(idx % C);
    long t = idx / C;
    int o = (int)(t % n);
    long w = t / n;
    int ow = o % ww, oh = (o / ww) % wh, od = o / (ww * wh);
    int gwi = (int)(w % gw), ghi = (int)((w / gw) % gh), gdi = (int)(w / ((long)gw * gh));
    int pd = (gdi * wd + od + sd) % Dp;
    int ph = (ghi * wh + oh + sh) % Hp;
    int pw = (gwi * ww + ow + sw) % Wp;
    float v = 0.f;
    if (pd < D && ph < H && pw < W)
        v = y[(((long)pd * H + ph) * W + pw) * C + c];
    xw[idx] = v;
}

// Reverse: scatter-add back (roll-back lands each token at its gather source).
__global__ void k_win_scatter(const float* __restrict__ xw, float* __restrict__ x,
                              int D, int H, int W, int C,
                              int wd, int wh, int ww,
                              int gd, int gh, int gw,
                              int sd, int sh, int sw,
                              int Dp, int Hp, int Wp)
{
    long n = (long)wd * wh * ww;
    long total = (long)gd * gh * gw * n * C;
    long idx = blockIdx.x * (long)blockDim.x + threadIdx.x;
    if (idx >= total) return;
    int c = (int)(idx % C);
    long t = idx / C;
    int o = (int)(t % n);
    long w = t / n;
    int ow = o % ww, oh = (o / ww) % wh, od = o / (ww * wh);
    int gwi = (int)(w % gw), ghi = (int)((w / gw) % gh), gdi = (int)(w / ((long)gw * gh));
    int pd = (gdi * wd + od + sd) % Dp;
    int ph = (ghi * wh + oh + sh) % Hp;
    int pw = (gwi * ww + ow + sw) % Wp;
    if (pd < D && ph < H && pw < W)
        x[(((long)pd * H + ph) * W + pw) * C + c] += xw[idx];
}

// ---------------------------------------------------------------------------
// Row-wise: add rel-pos bias (+ shift mask), softmax in place. One wave/row.
// ---------------------------------------------------------------------------
__global__ void __launch_bounds__(32)
k_attn_softmax(float* __restrict__ scores, const float* __restrict__ rpb,
               int n, int heads, int w0,
               int gd, int gh, int gw,
               int wd, int wh, int ww,
               int Dp, int Hp, int Wp,
               int sd, int sh, int sw, int masked)
{
    long r = blockIdx.x;
    int i = (int)(r % n);
    int h = (int)((r / n) % heads);
    long wl = r / ((long)n * heads);
    float* row = scores + ((wl * heads + h) * (long)n + i) * n;

    int wg  = w0 + (int)wl;
    int gwi = wg % gw, ghi = (wg / gw) % gh, gdi = wg / (gw * gh);
    int iw = i % ww, ih = (i / ww) % wh, id = i / (ww * wh);

    int ri = 0;
    if (masked) {
        int zd = gdi * wd + id, zh = ghi * wh + ih, zw = gwi * ww + iw;
        int ld = (zd < Dp - wd) ? 0 : ((zd < Dp - sd) ? 1 : 2);
        int lh = (zh < Hp - wh) ? 0 : ((zh < Hp - sh) ? 1 : 2);
        int lw = (zw < Wp - ww) ? 0 : ((zw < Wp - sw) ? 1 : 2);
        ri = (ld * 3 + lh) * 3 + lw;
    }

    int lane = threadIdx.x;
    float mx = -3.0e38f;
    for (int j = lane; j < n; j += 32) {
        int jw = j % ww, jh = (j / ww) % wh, jd = j / (ww * wh);
        int bidx = ((id - jd + wd - 1) * (2 * wh - 1) + (ih - jh + wh - 1)) * (2 * ww - 1)
                 + (iw - jw + ww - 1);
        float v = row[j] + rpb[(long)bidx * heads + h];
        if (masked) {
            int zd = gdi * wd + jd, zh = ghi * wh + jh, zw = gwi * ww + jw;
            int ld = (zd < Dp - wd) ? 0 : ((zd < Dp - sd) ? 1 : 2);
            int lh = (zh < Hp - wh) ? 0 : ((zh < Hp - sh) ? 1 : 2);
            int lw = (zw < Wp - ww) ? 0 : ((zw < Wp - sw) ? 1 : 2);
            if (((ld * 3 + lh) * 3 + lw) != ri) v -= 100.0f;
        }
        row[j] = v;
        mx = fmaxf(mx, v);
    }
    for (int o = 16; o; o >>= 1) mx = fmaxf(mx, __shfl_xor(mx, o));
    float sm = 0.f;
    for (int j = lane; j < n; j += 32) { float e = expf(row[j] - mx); row[j] = e; sm += e; }
    for (int o = 16; o; o >>= 1) sm += __shfl_xor(sm, o);
    float inv = 1.0f / sm;
    for (int j = lane; j < n; j += 32) row[j] *= inv;
}

// ---------------------------------------------------------------------------
// Small elementwise / glue kernels
// ---------------------------------------------------------------------------
__global__ void k_zero(float* __restrict__ p, long total)
{
    long i = blockIdx.x * (long)blockDim.x + threadIdx.x;
    if (i < total) p[i] = 0.f;
}

__global__ void k_gelu(float* __restrict__ x, long total)
{
    long i = blockIdx.x * (long)blockDim.x + threadIdx.x;
    if (i < total) {
        float v = x[i];
        x[i] = 0.5f * v * (1.0f + erff(v * 0.70710678118654752f));
    }
}

__global__ void k_im2col(const float* __restrict__ x, float* __restrict__ col)
{
    long total = 32768L * 64;
    long idx = blockIdx.x * (long)blockDim.x + threadIdx.x;
    if (idx >= total) return;
    int e = (int)(idx & 63);
    long v = idx >> 6;
    int pw = (int)(v % 32), ph = (int)((v / 32) % 32), pd = (int)(v / 1024);
    int k = e & 3, j = (e >> 2) & 3, i = e >> 4;
    col[idx] = x[((long)(pd * 4 + i) * 128 + (ph * 4 + j)) * 128 + (pw * 4 + k)];
}

// MONAI "merging" slice order (with duplicates): (d,h,w) offsets
__global__ void k_merge_gather(const float* __restrict__ x, float* __restrict__ y,
                               int D, int H, int W, int C)
{
    const int oi[8] = {0,1,0,0,1,0,0,1};
    const int oj[8] = {0,0,1,0,0,1,0,1};
    const int ok[8] = {0,0,0,1,1,0,1,1};
    int D2 = D / 2, H2 = H / 2, W2 = W / 2;
    long total = (long)D2 * H2 * W2 * 8 * C;
    long idx = blockIdx.x * (long)blockDim.x + threadIdx.x;
    if (idx >= total) return;
    int C8 = 8 * C;
    int cc = (int)(idx % C8);
    long t = idx / C8;
    int slot = cc / C, c = cc % C;
    int w2 = (int)(t % W2), h2 = (int)((t / W2) % H2), d2 = (int)(t / ((long)W2 * H2));
    long src = (((long)(2 * d2 + oi[slot]) * H + (2 * h2 + oj[slot])) * W
                + (2 * w2 + ok[slot])) * C + c;
    y[idx] = x[src];
}

__global__ void k_build_cls(const float* __restrict__ cls, const float* __restrict__ feat,
                            float* __restrict__ xcls)
{
    int i = blockIdx.x * blockDim.x + threadIdx.x;
    if (i < 768) xcls[i] = cls[i];
    else if (i < 9 * 768) xcls[i] = feat[i - 768];
}

// Classifier attention (single head, 9 tokens): scores out (pre-softmax), P@V
__global__ void k_head_attn(const float* __restrict__ qkvh,
                            float* __restrict__ scores_out, float* __restrict__ outh)
{
    __shared__ float sc[81];
    int t = threadIdx.x;
    const float scale = 1.0f / sqrtf(768.0f);
    if (t < 81) {
        int i = t / 9, j = t % 9;
        float s = 0.f;
        for (int d = 0; d < 768; ++d)
            s += qkvh[i * 2304 + d] * qkvh[j * 2304 + 768 + d];
        s *= scale;
        sc[t] = s;
        scores_out[t] = s;
    }
    __syncthreads();
    if (t < 9) {
        float mx = -3.0e38f;
        for (int j = 0; j < 9; ++j) mx = fmaxf(mx, sc[t * 9 + j]);
        float sm = 0.f;
        for (int j = 0; j < 9; ++j) { float e = expf(sc[t * 9 + j] - mx); sc[t * 9 + j] = e; sm += e; }
        for (int j = 0; j < 9; ++j) sc[t * 9 + j] /= sm;
    }
    __syncthreads();
    for (int idx = t; idx < 9 * 768; idx += blockDim.x) {
        int i = idx / 768, d = idx % 768;
        float s = 0.f;
        for (int j = 0; j < 9; ++j)
            s += sc[i * 9 + j] * qkvh[j * 2304 + 1536 + d];
        outh[idx] = s;
    }
}

__global__ void k_final(const float* __restrict__ proj0, const float* __restrict__ tw,
                        const float* __restrict__ tb, float* __restrict__ dout)
{
    __shared__ float red[256];
    float s = 0.f;
    for (int d = threadIdx.x; d < 768; d += 256) s += proj0[d] * tw[d];
    red[threadIdx.x] = s;
    __syncthreads();
    for (int o = 128; o; o >>= 1) {
        if ((int)threadIdx.x < o) red[threadIdx.x] += red[threadIdx.x + o];
        __syncthreads();
    }
    if (threadIdx.x == 0) {
        float logit = red[0] + tb[0];
        dout[1] = logit;
        dout[0] = (logit >= 0.f) ? 1.0f : 0.0f;   // sigmoid(x)>=0.5 <=> x>=0
    }
}

// ---------------------------------------------------------------------------
// Host orchestration
// ---------------------------------------------------------------------------
static const float* g_zpad_dev = nullptr;   // set per launch (points into d_ws)

static inline void gemm(hipStream_t st,
                        const float* A, long lda, long sA1, long sA2,
                        const float* B, long ldb, long sB1, long sB2, int transB,
                        float* C, long ldc, long sC1, long sC2,
                        int M, int N, int K, const float* bias, float alpha,
                        int accum, int b1, int b2)
{
    dim3 g((unsigned)((N + 15) / 16), (unsigned)((M + 15) / 16), (unsigned)(b1 * b2));
    k_gemm_wmma<<<g, dim3(32), 0, st>>>(A, B, C, bias, g_zpad_dev,
                                        M, N, K, lda, ldb, ldc,
                                        sA1, sA2, sB1, sB2, sC1, sC2,
                                        b2, alpha, transB, accum);
}

struct BlockP {
    const float *fc1_b, *fc1_w, *fc2_b, *fc2_w;
    const float *n1_b, *n1_g, *n2_b, *n2_g;
    const float *proj_b, *proj_w, *qkv_b, *qkv_w, *rpb;
};
struct LayerP { BlockP blk[2]; const float *mn_b, *mn_g, *mr_w; };

extern "C" void kernel_launch(void* const* d_in, const int* in_sizes, int n_in,
                              void* d_out, int out_size, void* d_ws, size_t ws_size,
                              hipStream_t stream)
{
    if (n_in < 126) return;
    hipStream_t st = stream;
    int ii = 0;
    auto nx = [&]() { return (const float*)d_in[ii++]; };

    // jax pytree order: sorted dict keys, "params" before "x"
    const float* attn_proj_b = nx();
    const float* attn_proj_w = nx();
    const float* attn_qkv_b  = nx();
    const float* attn_qkv_w  = nx();
    const float* cls_token   = nx();
    LayerP L[4];
    for (int i = 0; i < 4; ++i) {
        for (int j = 0; j < 2; ++j) {
            BlockP& b = L[i].blk[j];
            b.fc1_b = nx(); b.fc1_w = nx(); b.fc2_b = nx(); b.fc2_w = nx();
            b.n1_b  = nx(); b.n1_g  = nx(); b.n2_b  = nx(); b.n2_g  = nx();
            b.proj_b = nx(); b.proj_w = nx(); b.qkv_b = nx(); b.qkv_w = nx();
            b.rpb = nx();
        }
        L[i].mn_b = nx(); L[i].mn_g = nx(); L[i].mr_w = nx();
    }
    const float* patch_b = nx();
    const float* patch_w = nx();
    const float* tumor_b = nx();
    const float* tumor_w = nx();
    const float* xin     = nx();
    (void)in_sizes; (void)out_size;

    // workspace layout (floats)
    const long OFF_X = 0, OFF_Y = 1600000, OFF_XW = 3200000, OFF_ATTN = 5300000,
               OFF_QKV = 7400000, OFF_HID = 13600000, OFF_SC = 19900000,
               OFF_SMALL = 25600000, OFF_ZPAD = 25650000, TOTAL = 25700000;
    if (ws_size < (size_t)TOTAL * sizeof(float)) return;
    float* ws  = (float*)d_ws;
    float* xb  = ws + OFF_X;     // feature map, channel-last
    float* yb  = ws + OFF_Y;     // LN output
    float* xwb = ws + OFF_XW;    // windowed tokens / proj out
    float* atb = ws + OFF_ATTN;  // attention output
    float* qb  = ws + OFF_QKV;   // qkv
    float* hb  = ws + OFF_HID;   // MLP hidden / im2col / merge concat
    float* scb = ws + OFF_SC;    // attention scores chunk
    float* smallb = ws + OFF_SMALL;
    float* zpad = ws + OFF_ZPAD; // 4096+ zero floats for branchless GEMM edges
    g_zpad_dev = zpad;
    float* dout = (float*)d_out;

    k_zero<<<(4096 + 255) / 256, 256, 0, st>>>(zpad, 4096);

    // ---- patch embed (im2col + WMMA GEMM) ----
    {
        long tot = 32768L * 64;
        k_im2col<<<(unsigned)((tot + 255) / 256), 256, 0, st>>>(xin, hb);
        gemm(st, hb, 64, 0, 0, patch_w, 64, 0, 0, /*transB=*/1,
             xb, 48, 0, 0, 32768, 48, 64, patch_b, 1.0f, 0, 1, 1);
    }

    const int heads_arr[4] = {3, 6, 12, 24};
    int D = 32, H = 32, W = 32, C = 48;

    for (int li = 0; li < 4; ++li) {
        int heads = heads_arr[li];
        int wsz = (D <= 7) ? D : 7;
        int sft = (D <= 7) ? 0 : 3;
        int g1  = (D + wsz - 1) / wsz;
        int Dp  = g1 * wsz;
        int nW  = g1 * g1 * g1;
        int n   = wsz * wsz * wsz;
        int hd  = C / heads;
        int T   = D * H * W;
        int cw  = 48 / heads; if (cw < 1) cw = 1; if (cw > nW) cw = nW;
        float scale = 1.0f / sqrtf((float)hd);
        long tot = (long)nW * n * C;

        for (int bj = 0; bj < 2; ++bj) {
            const BlockP& bp = L[li].blk[bj];
            int shifted = (bj == 1 && sft > 0) ? 1 : 0;
            int s = shifted ? sft : 0;

            k_layernorm<<<T, 32, 0, st>>>(xb, yb, bp.n1_g, bp.n1_b, T, C);
            k_win_gather<<<(unsigned)((tot + 255) / 256), 256, 0, st>>>(
                yb, xwb, D, H, W, C, wsz, wsz, wsz, g1, g1, g1, s, s, s, Dp, Dp, Dp);

            gemm(st, xwb, C, 0, 0, bp.qkv_w, 3 * C, 0, 0, 0,
                 qb, 3 * C, 0, 0, nW * n, 3 * C, C, bp.qkv_b, 1.0f, 0, 1, 1);

            for (int w0 = 0; w0 < nW; w0 += cw) {
                int c2 = (nW - w0 < cw) ? (nW - w0) : cw;
                const float* qbase = qb + (long)w0 * n * 3 * C;
                // scores = scale * Q @ K^T   (batched over window x head)
                gemm(st, qbase, 3 * C, (long)n * 3 * C, hd,
                     qbase + C, 3 * C, (long)n * 3 * C, hd, /*transB=*/1,
                     scb, n, (long)heads * n * n, (long)n * n,
                     n, n, hd, nullptr, scale, 0, c2, heads);
                long rows = (long)c2 * heads * n;
                k_attn_softmax<<<(unsigned)rows, 32, 0, st>>>(
                    scb, bp.rpb, n, heads, w0, g1, g1, g1, wsz, wsz, wsz,
                    Dp, Dp, Dp, s, s, s, shifted);
                // out = P @ V  -> (token, head*hd) layout
                gemm(st, scb, n, (long)heads * n * n, (long)n * n,
                     qbase + 2 * C, 3 * C, (long)n * 3 * C, hd, 0,
                     atb + (long)w0 * n * C, C, (long)n * C, hd,
                     n, hd, n, nullptr, 1.0f, 0, c2, heads);
            }

            gemm(st, atb, C, 0, 0, bp.proj_w, C, 0, 0, 0,
                 xwb, C, 0, 0, nW * n, C, C, bp.proj_b, 1.0f, 0, 1, 1);
            k_win_scatter<<<(unsigned)((tot + 255) / 256), 256, 0, st>>>(
                xwb, xb, D, H, W, C, wsz, wsz, wsz, g1, g1, g1, s, s, s, Dp, Dp, Dp);

            // MLP: x += fc2(gelu(fc1(LN(x))))
            k_layernorm<<<T, 32, 0, st>>>(xb, yb, bp.n2_g, bp.n2_b, T, C);
            gemm(st, yb, C, 0, 0, bp.fc1_w, 4 * C, 0, 0, 0,
                 hb, 4 * C, 0, 0, T, 4 * C, C, bp.fc1_b, 1.0f, 0, 1, 1);
            long ht = (long)T * 4 * C;
            k_gelu<<<(unsigned)((ht + 255) / 256), 256, 0, st>>>(hb, ht);
            gemm(st, hb, 4 * C, 0, 0, bp.fc2_w, C, 0, 0, 0,
                 xb, C, 0, 0, T, C, 4 * C, bp.fc2_b, 1.0f, /*accum=*/1, 1, 1);
        }

        // patch merging (dims always even here)
        int D2 = D / 2;
        long T2 = (long)D2 * D2 * D2;
        long mt = T2 * 8 * C;
        k_merge_gather<<<(unsigned)((mt + 255) / 256), 256, 0, st>>>(xb, hb, D, H, W, C);
        k_layernorm<<<(unsigned)T2, 32, 0, st>>>(hb, yb, L[li].mn_g, L[li].mn_b, (int)T2, 8 * C);
        gemm(st, yb, 8 * C, 0, 0, L[li].mr_w, 2 * C, 0, 0, 0,
             xb, 2 * C, 0, 0, (int)T2, 2 * C, 8 * C, nullptr, 1.0f, 0, 1, 1);
        D = D2; H = D2; W = D2; C *= 2;
    }

    // proj_out: LN (no affine) over channels -> feat tokens (8 x 768)
    k_layernorm<<<8, 32, 0, st>>>(xb, yb, nullptr, nullptr, 8, 768);

    // classifier head
    float* xcls  = smallb;             // 9*768
    float* qkvh  = xcls + 9 * 768;     // 9*2304
    float* outh  = qkvh + 9 * 2304;    // 9*768
    float* projh = outh + 9 * 768;     // 9*768
    k_build_cls<<<(9 * 768 + 255) / 256, 256, 0, st>>>(cls_token, yb, xcls);
    gemm(st, xcls, 768, 0, 0, attn_qkv_w, 2304, 0, 0, 0,
         qkvh, 2304, 0, 0, 9, 2304, 768, attn_qkv_b, 1.0f, 0, 1, 1);
    k_head_attn<<<1, 128, 0, st>>>(qkvh, dout + 2, outh);
    gemm(st, outh, 768, 0, 0, attn_proj_w, 768, 0, 0, 0,
         projh, 768, 0, 0, 9, 768, 768, attn_proj_b, 1.0f, 0, 1, 1);
    k_final<<<1, 256, 0, st>>>(projh, tumor_w, tumor_b, dout);
}